// AttentionDecoder_25039659336089
// MI455X (gfx1250) — compile-verified
//
#include <hip/hip_runtime.h>
#include <math.h>

// ---------------------------------------------------------------------------
// Show-Attend-Tell decoder for MI455X (gfx1250, wave32).
// All GEMMs use V_WMMA_F32_16X16X4_F32 (pure fp32 matrix pipe, preserves the
// reference's fp32 numerics for K up to 1024). One wave owns a
// (16*MT) x (16*NT) macro-tile: A fragments are reused across NT column
// tiles, B fragments across MT row tiles, and a K-permutation makes every
// lane's K-data contiguous so fragments load as global_load_b128.
// global_prefetch_b8 hides L2 latency at low occupancy.
// ---------------------------------------------------------------------------

typedef __attribute__((ext_vector_type(2))) float v2f;
typedef __attribute__((ext_vector_type(4))) float v4f;
typedef __attribute__((ext_vector_type(8))) float v8f;

constexpr int Vc = 10000, Ec = 512, Hc = 512, Fc = 512, Ac = 512;
constexpr int Bc = 128, Lc = 196, Tc = 32, TDc = 31;

__device__ __forceinline__ float sigm(float x) { return 1.0f / (1.0f + __expf(-x)); }
__device__ __forceinline__ int declen_of(int len) { int d = len - 1; return d < 1 ? 1 : d; }

__device__ __forceinline__ v8f wmma4(v2f a, v2f b, v8f c) {
  return __builtin_amdgcn_wmma_f32_16x16x4_f32(false, a, false, b, (short)0, c,
                                               false, false);
}

// Accumulate a (16*MT) x (16*NT) macro-tile of  X(MxK) @ W^T, W row-major (N,K).
// K is processed in groups of 16. Within a group, WMMA s (s=0..3) covers
// global K = {k+2s, k+2s+1, k+8+2s, k+8+2s+1} (lane halves 0/1) — a valid
// permutation of the ISA chunk layout since A and B agree and K-sum commutes.
// Each lane then needs 8 contiguous floats per operand per group -> 2x b128.
template <int MT, int NT>
__device__ __forceinline__ void wmma_macro_acc(v8f (&acc)[MT][NT],
    const float* __restrict__ X, int ldx,
    const float* __restrict__ W, int ldw,
    int row0, int col0, int K)
{
  const int lane = threadIdx.x & 31;
  const int half = lane >> 4;
  const int l16  = lane & 15;
  const float* __restrict__ xr[MT];
#pragma unroll
  for (int i = 0; i < MT; ++i)
    xr[i] = X + (size_t)(row0 + 16 * i + l16) * ldx + 8 * half;
  const float* __restrict__ wr0 = W + (size_t)(col0 + l16) * ldw + 8 * half;

  for (int k = 0; k < K; k += 16) {
    v4f a0[MT], a1[MT];
#pragma unroll
    for (int i = 0; i < MT; ++i) {
      if (k + 16 < K) __builtin_prefetch(xr[i] + k + 16, 0, 1);
      a0[i] = *(const v4f*)(xr[i] + k);
      a1[i] = *(const v4f*)(xr[i] + k + 4);
    }
#pragma unroll
    for (int j = 0; j < NT; ++j) {
      const float* __restrict__ wr = wr0 + (size_t)(16 * j) * ldw;
      if (k + 16 < K) __builtin_prefetch(wr + k + 16, 0, 1);
      const v4f b0 = *(const v4f*)(wr + k);
      const v4f b1 = *(const v4f*)(wr + k + 4);
#pragma unroll
      for (int i = 0; i < MT; ++i) {
        v2f a, b;
        a.x = a0[i].x; a.y = a0[i].y; b.x = b0.x; b.y = b0.y;
        acc[i][j] = wmma4(a, b, acc[i][j]);
        a.x = a0[i].z; a.y = a0[i].w; b.x = b0.z; b.y = b0.w;
        acc[i][j] = wmma4(a, b, acc[i][j]);
        a.x = a1[i].x; a.y = a1[i].y; b.x = b1.x; b.y = b1.y;
        acc[i][j] = wmma4(a, b, acc[i][j]);
        a.x = a1[i].z; a.y = a1[i].w; b.x = b1.z; b.y = b1.w;
        acc[i][j] = wmma4(a, b, acc[i][j]);
      }
    }
  }
}

// act: 0 = none (+bias), 1 = tanh(+bias), 2 = sigmoid(+bias)
template <int MT, int NT>
__device__ __forceinline__ void wmma_macro_store(v8f (&acc)[MT][NT],
    float* __restrict__ C, int ldc, int row0, int col0,
    const float* __restrict__ bias, int act)
{
  const int lane = threadIdx.x & 31;
  const int half = lane >> 4;
  const int l16  = lane & 15;
#pragma unroll
  for (int j = 0; j < NT; ++j) {
    const int col = col0 + 16 * j + l16;
    const float bv = bias ? bias[col] : 0.0f;
#pragma unroll
    for (int i = 0; i < MT; ++i) {
#pragma unroll
      for (int v = 0; v < 8; ++v) {
        float val = acc[i][j][v] + bv;
        if (act == 1) val = tanhf(val);
        else if (act == 2) val = sigm(val);
        C[(size_t)(row0 + 16 * i + v + 8 * half) * ldc + col] = val;
      }
    }
  }
}

// ------------------------------ kernels ------------------------------------

__global__ void k_mean(const float* __restrict__ ann, float* __restrict__ mean_a) {
  int idx = blockIdx.x * blockDim.x + threadIdx.x;        // over B*F/4
  if (idx >= Bc * Fc / 4) return;
  int b = idx / (Fc / 4), f4 = idx % (Fc / 4);
  const float* p = ann + (size_t)b * Lc * Fc + 4 * f4;
  v4f s = {0.f, 0.f, 0.f, 0.f};
  for (int l = 0; l < Lc; ++l) s += *(const v4f*)(p + (size_t)l * Fc);
  *(v4f*)(mean_a + (size_t)b * Fc + 4 * f4) = s * (1.0f / (float)Lc);
}

// h0 = tanh(mean @ W_init_h^T + b), c0 likewise. 8 row-tiles x 8 col-groups x 2.
__global__ void k_init(const float* __restrict__ mean_a,
                       const float* __restrict__ Wh, const float* __restrict__ bh,
                       const float* __restrict__ Wc, const float* __restrict__ bc,
                       float* __restrict__ h, float* __restrict__ c) {
  int wave = (blockIdx.x * blockDim.x + threadIdx.x) >> 5;
  const int mt = Bc / 16, ng = Hc / 64, per = mt * ng;    // 8 x 8
  if (wave >= 2 * per) return;
  int which = wave / per, r = wave % per;
  int row0 = (r / ng) * 16, col0 = (r % ng) * 64;
  v8f acc[1][4] = {};
  wmma_macro_acc<1, 4>(acc, mean_a, Fc, which ? Wc : Wh, Fc, row0, col0, Fc);
  wmma_macro_store<1, 4>(acc, which ? c : h, Hc, row0, col0, which ? bc : bh, 1);
}

// ann_proj = annotations @ W_att_f^T + b_att_f  (25088 x 512, K=512), 32x64 tiles
__global__ void k_annproj(const float* __restrict__ ann, const float* __restrict__ Wf,
                          const float* __restrict__ bf, float* __restrict__ annproj) {
  int wave = (blockIdx.x * blockDim.x + threadIdx.x) >> 5;
  const int mt = (Bc * Lc) / 32, ng = Ac / 64;            // 784 x 8
  if (wave >= mt * ng) return;
  int row0 = (wave / ng) * 32, col0 = (wave % ng) * 64;
  v8f acc[2][4] = {};
  wmma_macro_acc<2, 4>(acc, ann, Fc, Wf, Fc, row0, col0, Fc);
  wmma_macro_store<2, 4>(acc, annproj, Ac, row0, col0, bf, 0);
}

// hp = h @ W_att_h^T + b_att_h ; gatepre = h @ W_beta^T + b_beta
__global__ void k_hp_gate(const float* __restrict__ h,
                          const float* __restrict__ Watt, const float* __restrict__ batt,
                          const float* __restrict__ Wbeta, const float* __restrict__ bbeta,
                          float* __restrict__ hp, float* __restrict__ gatepre) {
  int wave = (blockIdx.x * blockDim.x + threadIdx.x) >> 5;
  const int mt = Bc / 16, ng = Ac / 64, per = mt * ng;    // 8 x 8
  if (wave >= 2 * per) return;
  int which = wave / per, r = wave % per;
  int row0 = (r / ng) * 16, col0 = (r % ng) * 64;
  v8f acc[1][4] = {};
  wmma_macro_acc<1, 4>(acc, h, Hc, which ? Wbeta : Watt, Hc, row0, col0, Hc);
  wmma_macro_store<1, 4>(acc, which ? gatepre : hp, Ac, row0, col0,
                         which ? bbeta : batt, 0);
}

// Per-batch attention: scores -> softmax(L=196) -> context, gated context,
// embedding gather into x = [emb | gctx]. One block per batch row.
__global__ void k_attention(const float* __restrict__ annproj, const float* __restrict__ ann,
                            const float* __restrict__ hp, const float* __restrict__ gatepre,
                            const float* __restrict__ wv, const float* __restrict__ bv,
                            const float* __restrict__ Eemb, const int* __restrict__ captions,
                            const int* __restrict__ lengths,
                            float* __restrict__ x, float* __restrict__ alphas_out, int t) {
  __shared__ float s_alpha[256];
  __shared__ float s_red[256];
  const int b = blockIdx.x, tid = threadIdx.x;
  const bool active = declen_of(lengths[b]) > t;
  const int tok = captions[b * Tc + t];

  float sc = -3.4e38f;
  if (tid < Lc) {
    const v4f* ap  = (const v4f*)(annproj + ((size_t)b * Lc + tid) * Ac);
    const v4f* hb  = (const v4f*)(hp + (size_t)b * Ac);
    const v4f* wv4 = (const v4f*)wv;
    float acc = 0.f;
    for (int a = 0; a < Ac / 4; ++a) {
      v4f u = ap[a] + hb[a];
      v4f w = wv4[a];
      acc = fmaf(fmaxf(u.x, 0.f), w.x, acc);
      acc = fmaf(fmaxf(u.y, 0.f), w.y, acc);
      acc = fmaf(fmaxf(u.z, 0.f), w.z, acc);
      acc = fmaf(fmaxf(u.w, 0.f), w.w, acc);
    }
    sc = acc + bv[0];
  }
  s_red[tid] = sc; __syncthreads();
  for (int s = 128; s > 0; s >>= 1) {
    if (tid < s) s_red[tid] = fmaxf(s_red[tid], s_red[tid + s]);
    __syncthreads();
  }
  const float m = s_red[0];
  __syncthreads();
  float e = (tid < Lc) ? __expf(sc - m) : 0.f;
  s_red[tid] = e; __syncthreads();
  for (int s = 128; s > 0; s >>= 1) {
    if (tid < s) s_red[tid] += s_red[tid + s];
    __syncthreads();
  }
  const float inv = 1.0f / s_red[0];
  const float alpha = e * inv;
  s_alpha[tid] = alpha;
  __syncthreads();
  if (tid < Lc) alphas_out[((size_t)b * TDc + t) * Lc + tid] = active ? alpha : 0.f;

  if (tid < Fc / 4) {                                     // 128 threads, 4 f each
    v4f ctx = {0.f, 0.f, 0.f, 0.f};
    const float* ab = ann + (size_t)b * Lc * Fc + 4 * tid;
    for (int l = 0; l < Lc; ++l) {
      v4f v = *(const v4f*)(ab + (size_t)l * Fc);
      const float al = s_alpha[l];
      ctx.x = fmaf(al, v.x, ctx.x); ctx.y = fmaf(al, v.y, ctx.y);
      ctx.z = fmaf(al, v.z, ctx.z); ctx.w = fmaf(al, v.w, ctx.w);
    }
    const v4f gp = *(const v4f*)(gatepre + (size_t)b * Fc + 4 * tid);
    v4f g;
    g.x = sigm(gp.x) * ctx.x; g.y = sigm(gp.y) * ctx.y;
    g.z = sigm(gp.z) * ctx.z; g.w = sigm(gp.w) * ctx.w;
    *(v4f*)(x + (size_t)b * (Ec + Fc) + Ec + 4 * tid) = g;
    *(v4f*)(x + (size_t)b * (Ec + Fc) + 4 * tid) =
        *(const v4f*)(Eemb + (size_t)tok * Ec + 4 * tid);
  }
}

// gates = x @ W_ih^T (K=1024) + h @ W_hh^T (K=512) + b_ih + b_hh  (128 x 2048)
// 32x64 macro-tiles: 4 x 32 waves.
__global__ void k_gates(const float* __restrict__ x, const float* __restrict__ h,
                        const float* __restrict__ Wih, const float* __restrict__ Whh,
                        const float* __restrict__ bih, const float* __restrict__ bhh,
                        float* __restrict__ gates) {
  int wave = (blockIdx.x * blockDim.x + threadIdx.x) >> 5;
  const int mt = Bc / 32, ng = (4 * Hc) / 64;             // 4 x 32
  if (wave >= mt * ng) return;
  int row0 = (wave / ng) * 32, col0 = (wave % ng) * 64;
  v8f acc[2][4] = {};
  wmma_macro_acc<2, 4>(acc, x, Ec + Fc, Wih, Ec + Fc, row0, col0, Ec + Fc);
  wmma_macro_acc<2, 4>(acc, h, Hc, Whh, Hc, row0, col0, Hc);
  const int lane = threadIdx.x & 31;
  const int half = lane >> 4;
  const int l16  = lane & 15;
#pragma unroll
  for (int j = 0; j < 4; ++j) {
    const int col = col0 + 16 * j + l16;
    const float b2 = bih[col] + bhh[col];
#pragma unroll
    for (int i = 0; i < 2; ++i)
#pragma unroll
      for (int v = 0; v < 8; ++v)
        gates[(size_t)(row0 + 16 * i + v + 8 * half) * (4 * Hc) + col] =
            acc[i][j][v] + b2;
  }
}

// LSTM pointwise: c_new/h_new; masked in-place update of carried h, c.
__global__ void k_lstm_pw(const float* __restrict__ gates, const int* __restrict__ lengths,
                          float* __restrict__ h, float* __restrict__ c,
                          float* __restrict__ h_new, int t) {
  int idx = blockIdx.x * blockDim.x + threadIdx.x;
  if (idx >= Bc * Hc) return;
  int b = idx / Hc, j = idx % Hc;
  const float* g = gates + (size_t)b * 4 * Hc;
  float gi = g[j], gf = g[Hc + j], gg = g[2 * Hc + j], go = g[3 * Hc + j];
  float cn = sigm(gf) * c[idx] + sigm(gi) * tanhf(gg);
  float hn = sigm(go) * tanhf(cn);
  h_new[idx] = hn;                       // 'pre' uses h_new even when inactive
  if (declen_of(lengths[b]) > t) { c[idx] = cn; h[idx] = hn; }
}

// pre = tanh(emb@W_y^T + h_new@W_h^T + gctx@W_z^T)   (128 x 512)
__global__ void k_pre(const float* __restrict__ x, const float* __restrict__ h_new,
                      const float* __restrict__ Wy, const float* __restrict__ Wh,
                      const float* __restrict__ Wz, float* __restrict__ pre) {
  int wave = (blockIdx.x * blockDim.x + threadIdx.x) >> 5;
  const int mt = Bc / 16, ng = Ec / 64;                   // 8 x 8
  if (wave >= mt * ng) return;
  int row0 = (wave / ng) * 16, col0 = (wave % ng) * 64;
  v8f acc[1][4] = {};
  wmma_macro_acc<1, 4>(acc, x, Ec + Fc, Wy, Ec, row0, col0, Ec);       // emb slice
  wmma_macro_acc<1, 4>(acc, h_new, Hc, Wh, Hc, row0, col0, Hc);
  wmma_macro_acc<1, 4>(acc, x + Ec, Ec + Fc, Wz, Fc, row0, col0, Fc);  // gctx slice
  wmma_macro_store<1, 4>(acc, pre, Ec, row0, col0, nullptr, 1);
}

// logits = pre @ W_out^T + b_out, masked; strided store into (B, TD, V).
// 10000 = 125 groups of 5 column tiles; 32x80 macro-tile per wave (4 x 125).
__global__ void k_logits(const float* __restrict__ pre, const float* __restrict__ Wout,
                         const float* __restrict__ bout, const int* __restrict__ lengths,
                         float* __restrict__ out, int t) {
  int wave = (blockIdx.x * blockDim.x + threadIdx.x) >> 5;
  const int mt = Bc / 32, ng = Vc / 80;                   // 4 x 125
  if (wave >= mt * ng) return;
  int row0 = (wave / ng) * 32, col0 = (wave % ng) * 80;
  v8f acc[2][5] = {};
  wmma_macro_acc<2, 5>(acc, pre, Ec, Wout, Ec, row0, col0, Ec);
  const int lane = threadIdx.x & 31;
  const int half = lane >> 4;
  const int l16  = lane & 15;
#pragma unroll
  for (int j = 0; j < 5; ++j) {
    const int col = col0 + 16 * j + l16;
    const float bv = bout[col];
#pragma unroll
    for (int i = 0; i < 2; ++i)
#pragma unroll
      for (int v = 0; v < 8; ++v) {
        const int row = row0 + 16 * i + v + 8 * half;
        const bool active = declen_of(lengths[row]) > t;
        out[((size_t)row * TDc + t) * Vc + col] = active ? (acc[i][j][v] + bv) : 0.f;
      }
  }
}

__global__ void k_declen(const int* __restrict__ lengths, float* __restrict__ out) {
  int b = threadIdx.x;
  if (b < Bc) out[b] = (float)declen_of(lengths[b]);
}

// ------------------------------ launcher -----------------------------------

extern "C" void kernel_launch(void* const* d_in, const int* in_sizes, int n_in,
                              void* d_out, int out_size, void* d_ws, size_t ws_size,
                              hipStream_t stream) {
  (void)in_sizes; (void)n_in; (void)out_size; (void)ws_size;

  const float* annotations = (const float*)d_in[0];
  const int*   captions    = (const int*)  d_in[1];
  const int*   lengths     = (const int*)  d_in[2];
  const float* E_emb       = (const float*)d_in[3];
  const float* W_init_h    = (const float*)d_in[4];
  const float* b_init_h    = (const float*)d_in[5];
  const float* W_init_c    = (const float*)d_in[6];
  const float* b_init_c    = (const float*)d_in[7];
  const float* W_att_f     = (const float*)d_in[8];
  const float* b_att_f     = (const float*)d_in[9];
  const float* W_att_h     = (const float*)d_in[10];
  const float* b_att_h     = (const float*)d_in[11];
  const float* w_att_v     = (const float*)d_in[12];
  const float* b_att_v     = (const float*)d_in[13];
  const float* W_beta      = (const float*)d_in[14];
  const float* b_beta      = (const float*)d_in[15];
  const float* W_ih        = (const float*)d_in[16];
  const float* W_hh        = (const float*)d_in[17];
  const float* b_ih        = (const float*)d_in[18];
  const float* b_hh        = (const float*)d_in[19];
  const float* W_y         = (const float*)d_in[20];
  const float* W_h         = (const float*)d_in[21];
  const float* W_z         = (const float*)d_in[22];
  const float* W_out       = (const float*)d_in[23];
  const float* b_out       = (const float*)d_in[24];

  float* out        = (float*)d_out;
  float* logits_out = out;
  float* alphas_out = out + (size_t)Bc * TDc * Vc;
  float* declen_out = alphas_out + (size_t)Bc * TDc * Lc;

  float* ws = (float*)d_ws;
  size_t off = 0;
  float* annproj = ws + off; off += (size_t)Bc * Lc * Ac;   // 12.8M floats
  float* mean_a  = ws + off; off += (size_t)Bc * Fc;
  float* h       = ws + off; off += (size_t)Bc * Hc;
  float* c       = ws + off; off += (size_t)Bc * Hc;
  float* h_new   = ws + off; off += (size_t)Bc * Hc;
  float* hp      = ws + off; off += (size_t)Bc * Ac;
  float* gatepre = ws + off; off += (size_t)Bc * Fc;
  float* x       = ws + off; off += (size_t)Bc * (Ec + Fc);
  float* gates   = ws + off; off += (size_t)Bc * 4 * Hc;
  float* pre     = ws + off; off += (size_t)Bc * Ec;        // total ~13.7M floats

  // one-time precompute
  k_mean<<<(Bc * Fc / 4 + 255) / 256, 256, 0, stream>>>(annotations, mean_a);
  k_init<<<(2 * 8 * 8 * 32 + 255) / 256, 256, 0, stream>>>(
      mean_a, W_init_h, b_init_h, W_init_c, b_init_c, h, c);
  {
    const int waves = ((Bc * Lc) / 32) * (Ac / 64);         // 6272 macro-tiles
    k_annproj<<<(waves * 32 + 255) / 256, 256, 0, stream>>>(annotations, W_att_f,
                                                            b_att_f, annproj);
  }
  k_declen<<<1, 128, 0, stream>>>(lengths, declen_out);

  // sequential decode
  for (int t = 0; t < TDc; ++t) {
    k_hp_gate<<<(2 * 8 * 8 * 32 + 255) / 256, 256, 0, stream>>>(
        h, W_att_h, b_att_h, W_beta, b_beta, hp, gatepre);
    k_attention<<<Bc, 256, 0, stream>>>(annproj, annotations, hp, gatepre, w_att_v,
                                        b_att_v, E_emb, captions, lengths, x,
                                        alphas_out, t);
    k_gates<<<(4 * 32 * 32 + 255) / 256, 256, 0, stream>>>(x, h, W_ih, W_hh, b_ih,
                                                           b_hh, gates);
    k_lstm_pw<<<(Bc * Hc + 255) / 256, 256, 0, stream>>>(gates, lengths, h, c, h_new, t);
    k_pre<<<(8 * 8 * 32 + 255) / 256, 256, 0, stream>>>(x, h_new, W_y, W_h, W_z, pre);
    k_logits<<<(4 * 125 * 32 + 255) / 256, 256, 0, stream>>>(pre, W_out, b_out,
                                                             lengths, logits_out, t);
  }
}